// SS2D_65180423685500
// MI455X (gfx1250) — compile-verified
//
#include <hip/hip_runtime.h>
#include <math.h>

#define Bn 2
#define DM 192
#define HHH 64
#define WWW 64
#define DI 384
#define DS 16
#define ND 4
#define HW (HHH*WWW)   // 4096

typedef float v2f __attribute__((ext_vector_type(2)));
typedef float v8f __attribute__((ext_vector_type(8)));

__device__ __forceinline__ float fast_sigmoid(float x) {
  return __builtin_amdgcn_rcpf(1.f + __expf(-x));   // v_exp + v_rcp, no div chain
}

// ---------------- WMMA f32 16x16x4 fragment helpers (wave32) ----------------
// A: 16x4 row-major (lda): lane<16 -> {K0,K1} of row=lane ; lane>=16 -> {K2,K3}
// B: 4x16 row-major (ldb): lane<16 -> {K0,K1} of col=lane ; lane>=16 -> {K2,K3}
// C/D: vgpr i -> row = i + 8*(lane>=16), col = lane&15
__device__ __forceinline__ void wmma_store(float* __restrict__ C, long ldc,
                                           int lane, v8f acc) {
  const int half = lane >> 4;
  const int col  = lane & 15;
#pragma unroll
  for (int i = 0; i < 8; ++i)
    C[(long)(i + 8*half)*ldc + col] = acc[i];
}

// ---------------- Stage 1: in_proj (1x1) + SiLU ----------------
// Each wave: one 16-row M tile x four 16-col N tiles (4 indep accumulator chains,
// A fragment reused 4x per K-step).
__global__ void k_inproj(const float* __restrict__ W, const float* __restrict__ X,
                         float* __restrict__ O) {
  const int lane = threadIdx.x & 31;
  const int wv   = blockIdx.x * 8 + (threadIdx.x >> 5);
  const int b  = wv / (24 * 64);
  const int r  = wv % (24 * 64);
  const int mt = r / 64, ng = r % 64;
  const float* A  = W + mt * 16 * DM;
  const float* Bp = X + (size_t)b * DM * HW + ng * 64;
  const int half = lane >> 4, idx = lane & 15;
  v8f acc[4] = {};
#pragma unroll 4
  for (int k0 = 0; k0 < DM; k0 += 4) {
    const int ka = k0 + 2*half;
    v2f av = { A[idx*DM + ka], A[idx*DM + ka + 1] };
#pragma unroll
    for (int j = 0; j < 4; ++j) {
      v2f bv = { Bp[(size_t)ka*HW + j*16 + idx], Bp[(size_t)(ka+1)*HW + j*16 + idx] };
      acc[j] = __builtin_amdgcn_wmma_f32_16x16x4_f32(false, av, false, bv,
                                                     (short)0, acc[j], false, false);
    }
  }
  float* C = O + (size_t)b * DI * HW + (size_t)mt * 16 * HW + ng * 64;
#pragma unroll
  for (int j = 0; j < 4; ++j) {
#pragma unroll
    for (int i = 0; i < 8; ++i) {
      const float v = acc[j][i];
      acc[j][i] = v * fast_sigmoid(v);                // SiLU
    }
    wmma_store(C + j*16, HW, lane, acc[j]);
  }
}

// ---------------- Stage 2: depthwise 3x3 conv + bias ----------------
__global__ void k_dwconv(const float* __restrict__ in, const float* __restrict__ wt,
                         const float* __restrict__ bias, float* __restrict__ outb) {
  const int idx = blockIdx.x * 256 + threadIdx.x;
  const int xx = idx & 63;
  const int yy = (idx >> 6) & 63;
  const int c  = (idx >> 12) % DI;
  const int b  = idx / (DI * HW);
  const float* ibase = in + (size_t)(b * DI + c) * HW;
  float s = bias[c];
#pragma unroll
  for (int ky = 0; ky < 3; ++ky) {
    const int iy = yy + ky - 1;
    if (iy < 0 || iy >= HHH) continue;
#pragma unroll
    for (int kx = 0; kx < 3; ++kx) {
      const int ix = xx + kx - 1;
      if (ix < 0 || ix >= WWW) continue;
      s += ibase[iy * WWW + ix] * wt[c * 9 + ky * 3 + kx];
    }
  }
  outb[idx] = s;
}

// ---------------- Stage 3: x_down with TRANSPOSED store ----------------
// xp_t[b][w][h][s] so the scan can stage a whole column with coalesced float4s.
__global__ void k_xdown(const float* __restrict__ W, const float* __restrict__ H,
                        float* __restrict__ xp_t) {
  const int lane = threadIdx.x & 31;
  const int wv   = blockIdx.x * 8 + (threadIdx.x >> 5);   // 0..127
  const int b  = wv >> 6;
  const int ng = wv & 63;
  const float* Bp = H + (size_t)b * DI * HW + ng * 64;
  const int half = lane >> 4, idx = lane & 15;
  v8f acc[4] = {};
#pragma unroll 4
  for (int k0 = 0; k0 < DI; k0 += 4) {
    const int ka = k0 + 2*half;
    v2f av = { W[idx*DI + ka], W[idx*DI + ka + 1] };
#pragma unroll
    for (int j = 0; j < 4; ++j) {
      v2f bv = { Bp[(size_t)ka*HW + j*16 + idx], Bp[(size_t)(ka+1)*HW + j*16 + idx] };
      acc[j] = __builtin_amdgcn_wmma_f32_16x16x4_f32(false, av, false, bv,
                                                     (short)0, acc[j], false, false);
    }
  }
#pragma unroll
  for (int j = 0; j < 4; ++j) {
    const int n = ng * 64 + j * 16 + idx;                 // spatial index h*W+w
    float* dst = xp_t + ((size_t)(b * WWW + (n & 63)) * HHH + (n >> 6)) * DS + 8*half;
    float4 lo = { acc[j][0], acc[j][1], acc[j][2], acc[j][3] };
    float4 hi = { acc[j][4], acc[j][5], acc[j][6], acc[j][7] };
    *(float4*)dst       = lo;                             // s = 8*half .. +3
    *(float4*)(dst + 4) = hi;                             // s = 8*half+4 .. +7
  }
}

// ---------------- Stage 4: fused gates + 4-direction scan + merge ----------------
// WG per (b, cc); threads = (h 0..63) x (d 0..3). Double-buffered state: 2 barriers/step.
__global__ void k_scan(const float* __restrict__ hbuf,   // [B,DI,H,W]
                       const float* __restrict__ xp_t,   // [B,W,H,DS]
                       const float* __restrict__ w_up,   // [12*DI, DS]
                       const float* __restrict__ l_up,   // [4*DI, DS]
                       const float* __restrict__ u_up,   // [4*DI, DS]
                       const float* __restrict__ d_up,   // [4*DI, DS]
                       const float* __restrict__ m_w,    // [4]
                       float* __restrict__ y)            // [B,DI,H,W]
{
  const int tid = threadIdx.x;
  const int h = tid & 63;
  const int d = tid >> 6;
  const int b  = blockIdx.x / DI;
  const int cc = blockIdx.x % DI;
  const int c  = d * DI + cc;

  __shared__ float s_xp[HHH][DS + 1];        // padded: conflict-free K-dot reads
  __shared__ float s_state[2][ND][HHH];      // ping-pong recurrence state
  __shared__ float s_merge[ND][HHH];

  float w1[DS], w2[DS], w3[DS], wl[DS], wu[DS], wd[DS];
#pragma unroll
  for (int s = 0; s < DS; ++s) {
    w1[s] = w_up[(size_t)(c)               * DS + s];
    w2[s] = w_up[(size_t)(ND * DI + c)     * DS + s];
    w3[s] = w_up[(size_t)(2 * ND * DI + c) * DS + s];
    wl[s] = l_up[(size_t)c * DS + s];
    wu[s] = u_up[(size_t)c * DS + s];
    wd[s] = d_up[(size_t)c * DS + s];
  }
  const float mw = m_w[d];
  const float* hb = hbuf + (size_t)(b * DI + cc) * HW;

  s_state[0][d][h] = 0.f;

  for (int w = 0; w < WWW; ++w) {
    // stage column: 1024 contiguous floats -> one float4 per thread
    const float4 v4 = ((const float4*)(xp_t + (size_t)(b * WWW + w) * HHH * DS))[tid];
    const int h0 = tid >> 2, s0 = (tid & 3) << 2;
    s_xp[h0][s0 + 0] = v4.x; s_xp[h0][s0 + 1] = v4.y;
    s_xp[h0][s0 + 2] = v4.z; s_xp[h0][s0 + 3] = v4.w;
    __syncthreads();                                   // barrier A

    float a1 = 0, a2 = 0, a3 = 0, aL = 0, aU = 0, aD = 0;
#pragma unroll
    for (int s = 0; s < DS; ++s) {
      const float xv = s_xp[h][s];
      a1 += w1[s] * xv; a2 += w2[s] * xv; a3 += w3[s] * xv;
      aL += wl[s] * xv; aU += wu[s] * xv; aD += wd[s] * xv;
    }
    const float gl = fast_sigmoid(a1);
    const float gm = fast_sigmoid(a2);
    const float gr = fast_sigmoid(a3);
    float ssum;
    if (h == 0)            ssum = gm + gr;
    else if (h == HHH - 1) ssum = gl + gm;
    else                   ssum = gl + gm + gr;
    const float rs = __builtin_amdgcn_rcpf(fmaxf(ssum, 1e-7f));
    const float g1 = gl * rs, g2 = gm * rs, g3 = gr * rs;

    float xval;
    if      (d == 0) xval = hb[h * WWW + w];
    else if (d == 1) xval = hb[w * WWW + h];
    else if (d == 2) xval = hb[h * WWW + (WWW - 1 - w)];
    else             xval = hb[(WWW - 1 - w) * WWW + h];

    const float* stp = s_state[w & 1][d];
    const float up  = (h > 0)       ? stp[h - 1] : 0.f;
    const float mid = stp[h];
    const float dn  = (h < HHH - 1) ? stp[h + 1] : 0.f;
    const float Hn   = aL * xval + g1 * up + g2 * mid + g3 * dn;
    s_state[(w + 1) & 1][d][h] = Hn;                   // other buffer: no WAR hazard
    s_merge[d][h] = (Hn * aU + xval * aD) * mw;
    __syncthreads();                                   // barrier B
    if (d == 0)
      y[(size_t)(b * DI + cc) * HW + h * WWW + w] =
          s_merge[0][h] + s_merge[1][h] + s_merge[2][h] + s_merge[3][h];
  }
}

// ---------------- Stage 5a: LayerNorm over channels ----------------
__global__ void k_ln(const float* __restrict__ y, const float* __restrict__ nw,
                     const float* __restrict__ nb, float* __restrict__ yln) {
  __shared__ float sh1[64], sh2[64];
  const int tid = threadIdx.x;
  const int p = blockIdx.x;
  const int b = p >> 12, hw = p & 4095;
  const float* base = y + (size_t)b * DI * HW + hw;
  float s = 0.f, s2 = 0.f;
#pragma unroll
  for (int c = tid; c < DI; c += 64) {
    const float v = base[(size_t)c * HW];
    s += v; s2 += v * v;
  }
  sh1[tid] = s; sh2[tid] = s2;
  __syncthreads();
  for (int off = 32; off > 0; off >>= 1) {
    if (tid < off) { sh1[tid] += sh1[tid + off]; sh2[tid] += sh2[tid + off]; }
    __syncthreads();
  }
  const float mu  = sh1[0] * (1.f / DI);
  const float var = sh2[0] * (1.f / DI) - mu * mu;
  const float inv = rsqrtf(var + 1e-5f);
  float* obase = yln + (size_t)b * DI * HW + hw;
#pragma unroll
  for (int c = tid; c < DI; c += 64) {
    const float v = base[(size_t)c * HW];
    obase[(size_t)c * HW] = (v - mu) * inv * nw[c] + nb[c];
  }
}

// ---------------- Stage 5b: GRN Gx = sqrt(sum_{hw} y^2) per (b,c) ----------------
__global__ void k_gx(const float* __restrict__ yln, float* __restrict__ Gx) {
  __shared__ float sh[256];
  const int tid = threadIdx.x;
  const float* base = yln + (size_t)blockIdx.x * HW;
  float s = 0.f;
  for (int i = tid; i < HW; i += 256) { const float v = base[i]; s += v * v; }
  sh[tid] = s; __syncthreads();
  for (int off = 128; off > 0; off >>= 1) {
    if (tid < off) sh[tid] += sh[tid + off];
    __syncthreads();
  }
  if (tid == 0) Gx[blockIdx.x] = sqrtf(sh[0]);
}

// ---------------- Stage 5c: scale[b,c] = 1 + gamma[c]*Gx/(mean_c(Gx)+1e-6) -------
__global__ void k_scale(const float* __restrict__ Gx, const float* __restrict__ gamma,
                        float* __restrict__ scale) {
  __shared__ float sh[128];
  const int tid = threadIdx.x;
  const int b = blockIdx.x;
  const float g0 = Gx[b * DI + tid];
  const float g1 = Gx[b * DI + tid + 128];
  const float g2 = Gx[b * DI + tid + 256];
  sh[tid] = g0 + g1 + g2;
  __syncthreads();
  for (int off = 64; off > 0; off >>= 1) {
    if (tid < off) sh[tid] += sh[tid + off];
    __syncthreads();
  }
  const float inv = 1.f / (sh[0] * (1.f / DI) + 1e-6f);
  scale[b * DI + tid]       = 1.f + gamma[tid]       * g0 * inv;
  scale[b * DI + tid + 128] = 1.f + gamma[tid + 128] * g1 * inv;
  scale[b * DI + tid + 256] = 1.f + gamma[tid + 256] * g2 * inv;
}

// ---------------- Stage 6: out_proj with fused GRN epilogue on B-load ------------
// out[b,DM,HW] = W[DM,DI] @ (yln * scale[b,c] + beta[c]) ; 4 N-tiles per wave
__global__ void k_outproj(const float* __restrict__ W, const float* __restrict__ yln,
                          const float* __restrict__ scale, const float* __restrict__ beta,
                          float* __restrict__ out) {
  const int lane = threadIdx.x & 31;
  const int wv   = blockIdx.x * 8 + (threadIdx.x >> 5);
  const int b  = wv / (12 * 64);
  const int r  = wv % (12 * 64);
  const int mt = r / 64, ng = r % 64;
  const float* A  = W + mt * 16 * DI;
  const float* Bp = yln + (size_t)b * DI * HW + ng * 64;
  const float* sc = scale + b * DI;
  const int half = lane >> 4, idx = lane & 15;
  v8f acc[4] = {};
#pragma unroll 4
  for (int k0 = 0; k0 < DI; k0 += 4) {
    const int ka = k0 + 2 * half;
    v2f av = { A[idx * DI + ka], A[idx * DI + ka + 1] };
    const float sc0 = sc[ka], sc1 = sc[ka + 1];
    const float bt0 = beta[ka], bt1 = beta[ka + 1];
#pragma unroll
    for (int j = 0; j < 4; ++j) {
      v2f bv = { Bp[(size_t)ka * HW + j*16 + idx]       * sc0 + bt0,
                 Bp[(size_t)(ka + 1) * HW + j*16 + idx] * sc1 + bt1 };
      acc[j] = __builtin_amdgcn_wmma_f32_16x16x4_f32(false, av, false, bv,
                                                     (short)0, acc[j], false, false);
    }
  }
  float* C = out + (size_t)b * DM * HW + (size_t)mt * 16 * HW + ng * 64;
#pragma unroll
  for (int j = 0; j < 4; ++j)
    wmma_store(C + j * 16, HW, lane, acc[j]);
}

// ---------------- Launcher ----------------
extern "C" void kernel_launch(void* const* d_in, const int* in_sizes, int n_in,
                              void* d_out, int out_size, void* d_ws, size_t ws_size,
                              hipStream_t stream) {
  const float* x       = (const float*)d_in[0];
  const float* in_proj = (const float*)d_in[1];
  const float* dwc_w   = (const float*)d_in[2];
  const float* dwc_b   = (const float*)d_in[3];
  const float* xdown_w = (const float*)d_in[4];
  const float* w_up    = (const float*)d_in[5];
  const float* l_up    = (const float*)d_in[6];
  const float* u_up    = (const float*)d_in[7];
  const float* d_up    = (const float*)d_in[8];
  const float* m_w     = (const float*)d_in[9];
  const float* grn_g   = (const float*)d_in[10];
  const float* grn_b   = (const float*)d_in[11];
  const float* norm_w  = (const float*)d_in[12];
  const float* norm_b  = (const float*)d_in[13];
  const float* outp_w  = (const float*)d_in[14];
  float* out = (float*)d_out;

  float* ws    = (float*)d_ws;
  float* hpre  = ws;                                  // [B,DI,HW]
  float* hbuf  = hpre + (size_t)Bn * DI * HW;         // [B,DI,HW]
  float* xp_t  = hbuf + (size_t)Bn * DI * HW;         // [B,W,H,DS]
  float* ybuf  = xp_t + (size_t)Bn * DS * HW;         // [B,DI,HW]
  float* yln   = ybuf + (size_t)Bn * DI * HW;         // [B,DI,HW]
  float* Gx    = yln  + (size_t)Bn * DI * HW;         // [B*DI]
  float* scl   = Gx   + Bn * DI;                      // [B*DI]

  k_inproj <<<384, 256, 0, stream>>>(in_proj, x, hpre);           // 2*24*64 waves
  k_dwconv <<<(Bn * DI * HW) / 256, 256, 0, stream>>>(hpre, dwc_w, dwc_b, hbuf);
  k_xdown  <<<16, 256, 0, stream>>>(xdown_w, hbuf, xp_t);         // 2*64 waves
  k_scan   <<<Bn * DI, 256, 0, stream>>>(hbuf, xp_t, w_up, l_up, u_up, d_up, m_w, ybuf);
  k_ln     <<<Bn * HW, 64, 0, stream>>>(ybuf, norm_w, norm_b, yln);
  k_gx     <<<Bn * DI, 256, 0, stream>>>(yln, Gx);
  k_scale  <<<Bn, 128, 0, stream>>>(Gx, grn_g, scl);
  k_outproj<<<192, 256, 0, stream>>>(outp_w, yln, scl, grn_b, out); // 2*12*64 waves
}